// CapGATattentionGRU_79302276153724
// MI455X (gfx1250) — compile-verified
//
#include <hip/hip_runtime.h>
#include <math.h>

typedef float v2f __attribute__((ext_vector_type(2)));
typedef float v8f __attribute__((ext_vector_type(8)));

#define FDIM 2048
#define GDIM 6144          // 3 * FDIM (r,z,n gates stacked)
#define TDIM 12
#define NNODE 128
#define HDIM 16
#define KCHUNKS 4
#define KLEN (FDIM / KCHUNKS)   // 512
#define NTILES (GDIM / 16)      // 384

__device__ __forceinline__ float sigmoidf_(float x) {
    return 1.0f / (1.0f + __expf(-x));
}

// ---------------------------------------------------------------------------
// Kernel 0: build padded (16, 2048) x with nan_to_num(nan=0, posinf=1.0,
// neginf=-maxfloat); rows 12..15 zero.
// ---------------------------------------------------------------------------
__global__ void prep_x_kernel(const float* __restrict__ in, float* __restrict__ xpad) {
    int idx = blockIdx.x * blockDim.x + threadIdx.x;
    if (idx >= 16 * FDIM) return;
    int row = idx >> 11;
    int f   = idx & (FDIM - 1);
    float v = 0.0f;
    if (row < TDIM) {
        v = in[row * FDIM + f];
        if (v != v) v = 0.0f;                       // NaN -> 0
        else if (isinf(v)) v = (v > 0.0f) ? 1.0f : -3.4028234663852886e38f;
    }
    xpad[idx] = v;
}

// ---------------------------------------------------------------------------
// GRU input GEMM, split-K partials:
//   gi = x(16x2048) @ W_ih^T(2048x6144)  via V_WMMA_F32_16X16X4_F32
// Wave (ntile, kchunk): 16-column tile of gi over a K-chunk of 512.
// 384 N-tiles x 4 K-chunks = 1536 waves -> enough outstanding loads to fill
// the 23.3 TB/s latency-bandwidth product (the kernel is pure weight
// streaming; each W element is read exactly once).
// Partials stored as part[(kc*16 + m)*6144 + n], reduced in the epilogue.
// ---------------------------------------------------------------------------
__global__ void gru_gemm_partial_kernel(const float* __restrict__ x,     // (16,2048) padded
                                        const float* __restrict__ W_ih,  // (6144,2048)
                                        float* __restrict__ part) {      // (4,16,6144)
    int wave = (blockIdx.x * blockDim.x + threadIdx.x) >> 5;
    if (wave >= NTILES * KCHUNKS) return;       // uniform per wave
    int lane = threadIdx.x & 31;
    int half = lane >> 4;                        // 0: K pair {0,1}, 1: K pair {2,3}
    int l16  = lane & 15;
    int ntile = wave >> 2;
    int kc    = wave & (KCHUNKS - 1);
    int nbase = ntile * 16;
    int kbase = kc * KLEN;

    // A fragment: row m = l16, contiguous float2 at k + 2*half
    const float* ap = x + (size_t)l16 * FDIM + kbase + 2 * half;
    // B fragment: column n = l16 of B(K x 16) == row (nbase + l16) of W_ih
    const float* bp = W_ih + (size_t)(nbase + l16) * FDIM + kbase + 2 * half;

    v8f acc = {};
    for (int k0 = 0; k0 < KLEN; k0 += 32) {
        if (k0 + 256 < KLEN)
            __builtin_prefetch((const void*)(bp + k0 + 256), 0, 0);
#pragma unroll
        for (int kk = 0; kk < 32; kk += 4) {
            v2f a = *(const v2f*)(ap + k0 + kk);
            v2f b = *(const v2f*)(bp + k0 + kk);
            acc = __builtin_amdgcn_wmma_f32_16x16x4_f32(false, a, false, b, (short)0, acc, false, false);
        }
    }

    // C/D layout: VGPR v, lane -> (M = v + 8*half, N = l16)
#pragma unroll
    for (int v = 0; v < 8; v++) {
        int m = v + 8 * half;
        part[(size_t)(kc * 16 + m) * GDIM + nbase + l16] = acc[v];
    }
}

// ---------------------------------------------------------------------------
// Split-K reduction + GRU gate epilogue. One GRU step with h0 == 0 (the
// reference scans axis 0 which has length 1), so gh = b_hh exactly and W_hh
// is never read (exact, not an approximation).
//   r = sig(gi_r + bih_r + bhh_r); z = sig(gi_z + bih_z + bhh_z)
//   n = tanh(gi_n + bih_n + r*bhh_n); h = (1-z)*n   [+ z*h0 = 0]
// ---------------------------------------------------------------------------
__global__ void gru_epilogue_kernel(const float* __restrict__ part,  // (4,16,6144)
                                    const float* __restrict__ b_ih,  // (6144)
                                    const float* __restrict__ b_hh,  // (6144)
                                    float* __restrict__ hout,        // (16,2048)
                                    int apply_relu) {
    int idx = blockIdx.x * blockDim.x + threadIdx.x;
    if (idx >= 16 * FDIM) return;
    int m   = idx >> 11;
    int col = idx & (FDIM - 1);
    float gr = 0.0f, gz = 0.0f, gn = 0.0f;
#pragma unroll
    for (int kc = 0; kc < KCHUNKS; kc++) {
        const float* p = part + (size_t)(kc * 16 + m) * GDIM;
        gr += p[col];
        gz += p[FDIM + col];
        gn += p[2 * FDIM + col];
    }
    float r = sigmoidf_(gr + b_ih[col]            + b_hh[col]);
    float z = sigmoidf_(gz + b_ih[FDIM + col]     + b_hh[FDIM + col]);
    float n = tanhf(gn + b_ih[2 * FDIM + col] + r * b_hh[2 * FDIM + col]);
    float h = (1.0f - z) * n;                     // h_prev == 0
    if (apply_relu) h = fmaxf(h, 0.0f);
    hout[idx] = h;
}

// ---------------------------------------------------------------------------
// Attention over time. One thread per feature f (2048 threads).
// aw[f,t'] = b_att[t'] + sum_t emb[t,f] * W_att[t',t]; softmax over t';
// att[f] = tanh(sum_t' p[t'] * emb[t',f]).
// ---------------------------------------------------------------------------
__global__ void att_kernel(const float* __restrict__ emb, const float* __restrict__ W_att,
                           const float* __restrict__ b_att, float* __restrict__ xg) {
    int f = blockIdx.x * blockDim.x + threadIdx.x;
    if (f >= FDIM) return;
    float e[TDIM];
#pragma unroll
    for (int t = 0; t < TDIM; t++) e[t] = emb[t * FDIM + f];
    float aw[TDIM];
    float mx = -1e30f;
#pragma unroll
    for (int tp = 0; tp < TDIM; tp++) {
        float s = b_att[tp];
#pragma unroll
        for (int t = 0; t < TDIM; t++) s += e[t] * W_att[tp * TDIM + t];
        aw[tp] = s;
        mx = fmaxf(mx, s);
    }
    float ssum = 0.0f, acc = 0.0f;
#pragma unroll
    for (int tp = 0; tp < TDIM; tp++) {
        float w = __expf(aw[tp] - mx);
        ssum += w;
        acc += w * e[tp];
    }
    xg[f] = tanhf(acc / ssum);
}

// ---------------------------------------------------------------------------
// GATv2 layer (dense graph, 128 nodes, heads=1) + ReLU.
// One block, 128 threads; thread i owns target node i.
// ---------------------------------------------------------------------------
__global__ void gat_kernel(const float* __restrict__ xin,
                           const float* __restrict__ Wl, const float* __restrict__ bl,
                           const float* __restrict__ Wr, const float* __restrict__ br,
                           const float* __restrict__ a,  const float* __restrict__ gb,
                           float* __restrict__ gout) {
    __shared__ float xl[NNODE * HDIM];
    __shared__ float xr[NNODE * HDIM];
    __shared__ float av[HDIM];
    int i = threadIdx.x;
    if (i < HDIM) av[i] = a[i];
    float xi[HDIM];
#pragma unroll
    for (int h = 0; h < HDIM; h++) xi[h] = xin[i * HDIM + h];
#pragma unroll
    for (int h = 0; h < HDIM; h++) {
        float sl = bl[h], sr = br[h];
#pragma unroll
        for (int k = 0; k < HDIM; k++) {
            sl += xi[k] * Wl[h * HDIM + k];
            sr += xi[k] * Wr[h * HDIM + k];
        }
        xl[i * HDIM + h] = sl;
        xr[i * HDIM + h] = sr;
    }
    __syncthreads();
    float xri[HDIM];
#pragma unroll
    for (int h = 0; h < HDIM; h++) xri[h] = xr[i * HDIM + h];
    float smax = -1e30f;
    for (int j = 0; j < NNODE; j++) {
        float s = 0.0f;
#pragma unroll
        for (int h = 0; h < HDIM; h++) {
            float t = xri[h] + xl[j * HDIM + h];
            t = (t > 0.0f) ? t : 0.2f * t;        // leaky_relu 0.2
            s += t * av[h];
        }
        smax = fmaxf(smax, s);
    }
    float ssum = 0.0f;
    float acc[HDIM];
#pragma unroll
    for (int h = 0; h < HDIM; h++) acc[h] = 0.0f;
    for (int j = 0; j < NNODE; j++) {
        float s = 0.0f;
#pragma unroll
        for (int h = 0; h < HDIM; h++) {
            float t = xri[h] + xl[j * HDIM + h];
            t = (t > 0.0f) ? t : 0.2f * t;
            s += t * av[h];
        }
        float w = __expf(s - smax);
        ssum += w;
#pragma unroll
        for (int h = 0; h < HDIM; h++) acc[h] += w * xl[j * HDIM + h];
    }
    float inv = 1.0f / ssum;
#pragma unroll
    for (int h = 0; h < HDIM; h++)
        gout[i * HDIM + h] = fmaxf(acc[h] * inv + gb[h], 0.0f);
}

// ---------------------------------------------------------------------------
// fusion_vec = concat(att_vec, g0+g1) -> (128, 32)
// ---------------------------------------------------------------------------
__global__ void fuse_kernel(const float* __restrict__ xg, const float* __restrict__ g0,
                            const float* __restrict__ g1, float* __restrict__ fusion) {
    int idx = blockIdx.x * blockDim.x + threadIdx.x;
    if (idx >= NNODE * 32) return;
    int c = idx >> 5, i = idx & 31;
    float v;
    if (i < HDIM) v = xg[c * HDIM + i];
    else          v = g0[c * HDIM + (i - HDIM)] + g1[c * HDIM + (i - HDIM)];
    fusion[idx] = v;
}

// ---------------------------------------------------------------------------
// priors[o,c,l] = sum_i W_caps[o,l,i] * fusion[c,i]
// GEMM M=2048 (o*16+l), N=128 (c), K=32 via V_WMMA_F32_16X16X4_F32.
// Stored as priors[(o*16+l)*128 + c].
// ---------------------------------------------------------------------------
__global__ void priors_kernel(const float* __restrict__ Wcaps,   // (2048, 32) row-major
                              const float* __restrict__ fusion,  // (128, 32) row-major
                              float* __restrict__ priors) {      // (2048, 128)
    int wave = (blockIdx.x * blockDim.x + threadIdx.x) >> 5;
    if (wave >= 128 * 8) return;
    int lane = threadIdx.x & 31;
    int half = lane >> 4;
    int l16  = lane & 15;
    int mt = wave >> 3, nt = wave & 7;
    int mbase = mt * 16, cbase = nt * 16;
    const float* ap = Wcaps  + (size_t)(mbase + l16) * 32 + 2 * half;
    const float* bp = fusion + (size_t)(cbase + l16) * 32 + 2 * half;
    v8f acc = {};
#pragma unroll
    for (int k0 = 0; k0 < 32; k0 += 4) {
        v2f a = *(const v2f*)(ap + k0);
        v2f b = *(const v2f*)(bp + k0);
        acc = __builtin_amdgcn_wmma_f32_16x16x4_f32(false, a, false, b, (short)0, acc, false, false);
    }
#pragma unroll
    for (int v = 0; v < 8; v++) {
        int M = mbase + v + 8 * half;
        priors[(size_t)M * NNODE + cbase + l16] = acc[v];
    }
}

// ---------------------------------------------------------------------------
// Dynamic routing (3 iters) + final tanh(relu(out) @ W_fus^T + b).
// Single block, 256 threads. logits/p in global ws, capsules in LDS.
// ---------------------------------------------------------------------------
__global__ void routing_kernel(const float* __restrict__ priors,  // (2048,128)
                               const float* __restrict__ W_fus,   // (128,16)
                               const float* __restrict__ b_fus,   // (128)
                               float* __restrict__ logits,        // (128,128) ws
                               float* __restrict__ p,             // (128,128) ws
                               float* __restrict__ out) {         // (128,128) final
    __shared__ float oc[NNODE * HDIM];   // out capsules (128,16)
    int tid = threadIdx.x;

    for (int t = tid; t < NNODE * NNODE; t += blockDim.x) logits[t] = 0.0f;
    __syncthreads();

    for (int r = 0; r < 3; r++) {
        // softmax over o (axis=1) for each in-capsule c
        if (tid < NNODE) {
            int c = tid;
            float mx = -1e30f;
            for (int o = 0; o < NNODE; o++) mx = fmaxf(mx, logits[o * NNODE + c]);
            float s = 0.0f;
            for (int o = 0; o < NNODE; o++) s += __expf(logits[o * NNODE + c] - mx);
            float inv = 1.0f / s;
            for (int o = 0; o < NNODE; o++)
                p[o * NNODE + c] = __expf(logits[o * NNODE + c] - mx) * inv;
        }
        __syncthreads();
        // out[o,l] = sum_c p[o,c] * priors[o,c,l]
        for (int t = tid; t < NNODE * HDIM; t += blockDim.x) {
            int o = t >> 4, l = t & 15;
            const float* pr = priors + (size_t)(o * HDIM + l) * NNODE;
            const float* pp = p + (size_t)o * NNODE;
            float s = 0.0f;
            for (int c = 0; c < NNODE; c++) s += pp[c] * pr[c];
            oc[t] = s;
        }
        __syncthreads();
        // squash per out-capsule
        if (tid < NNODE) {
            int o = tid;
            float n2 = 0.0f;
            for (int l = 0; l < HDIM; l++) { float v = oc[o * HDIM + l]; n2 += v * v; }
            float scale = n2 / ((1.0f + n2) * sqrtf(n2 + 1e-8f));
            for (int l = 0; l < HDIM; l++) oc[o * HDIM + l] *= scale;
        }
        __syncthreads();
        if (r < 2) {
            // logits[o,c] += sum_l priors[o,c,l] * out[o,l]
            for (int t = tid; t < NNODE * NNODE; t += blockDim.x) {
                int o = t >> 7, c = t & 127;
                float s = 0.0f;
                for (int l = 0; l < HDIM; l++)
                    s += priors[(size_t)(o * HDIM + l) * NNODE + c] * oc[o * HDIM + l];
                logits[t] += s;
            }
            __syncthreads();
        }
    }
    // final: out[o,j] = tanh(b_fus[j] + sum_l relu(oc[o,l]) * W_fus[j,l])
    for (int t = tid; t < NNODE * NNODE; t += blockDim.x) {
        int o = t >> 7, j = t & 127;
        float s = b_fus[j];
        for (int l = 0; l < HDIM; l++)
            s += fmaxf(oc[o * HDIM + l], 0.0f) * W_fus[j * HDIM + l];
        out[t] = tanhf(s);
    }
}

// ---------------------------------------------------------------------------
extern "C" void kernel_launch(void* const* d_in, const int* in_sizes, int n_in,
                              void* d_out, int out_size, void* d_ws, size_t ws_size,
                              hipStream_t stream) {
    const float* inputs = (const float*)d_in[0];
    const float* W_ih0  = (const float*)d_in[1];
    // d_in[2] = W_hh0 : multiplied by h0 == 0, exactly unused
    const float* b_ih0  = (const float*)d_in[3];
    const float* b_hh0  = (const float*)d_in[4];
    const float* W_ih1  = (const float*)d_in[5];
    // d_in[6] = W_hh1 : exactly unused
    const float* b_ih1  = (const float*)d_in[7];
    const float* b_hh1  = (const float*)d_in[8];
    const float* W_att  = (const float*)d_in[9];
    const float* b_att  = (const float*)d_in[10];
    const float* Wl0 = (const float*)d_in[11]; const float* bl0 = (const float*)d_in[12];
    const float* Wr0 = (const float*)d_in[13]; const float* br0 = (const float*)d_in[14];
    const float* a0  = (const float*)d_in[15]; const float* gb0 = (const float*)d_in[16];
    const float* Wl1 = (const float*)d_in[17]; const float* bl1 = (const float*)d_in[18];
    const float* Wr1 = (const float*)d_in[19]; const float* br1 = (const float*)d_in[20];
    const float* a1  = (const float*)d_in[21]; const float* gb1 = (const float*)d_in[22];
    const float* W_caps = (const float*)d_in[23];
    const float* W_fus  = (const float*)d_in[24];
    const float* b_fus  = (const float*)d_in[25];
    float* outp = (float*)d_out;

    // workspace layout (floats)
    float* ws = (float*)d_ws;
    float* x_pad   = ws;                               // 16*2048
    float* h0_pad  = x_pad   + 16 * FDIM;              // 16*2048
    float* emb_pad = h0_pad  + 16 * FDIM;              // 16*2048
    float* xg      = emb_pad + 16 * FDIM;              // 2048
    float* g0      = xg      + FDIM;                   // 2048
    float* g1      = g0      + NNODE * HDIM;           // 2048
    float* fusion  = g1      + NNODE * HDIM;           // 128*32
    float* priors  = fusion  + NNODE * 32;             // 2048*128
    float* logits  = priors  + 2048 * NNODE;           // 128*128
    float* pbuf    = logits  + NNODE * NNODE;          // 128*128
    float* part    = pbuf    + NNODE * NNODE;          // 4*16*6144

    prep_x_kernel<<<(16 * FDIM + 255) / 256, 256, 0, stream>>>(inputs, x_pad);

    // GRU layer 0: 1536 partial-GEMM waves (8 waves/block) + epilogue
    gru_gemm_partial_kernel<<<NTILES * KCHUNKS / 8, 256, 0, stream>>>(x_pad, W_ih0, part);
    gru_epilogue_kernel<<<(16 * FDIM + 255) / 256, 256, 0, stream>>>(part, b_ih0, b_hh0, h0_pad, 0);

    // GRU layer 1 (+ ReLU -> emb)
    gru_gemm_partial_kernel<<<NTILES * KCHUNKS / 8, 256, 0, stream>>>(h0_pad, W_ih1, part);
    gru_epilogue_kernel<<<(16 * FDIM + 255) / 256, 256, 0, stream>>>(part, b_ih1, b_hh1, emb_pad, 1);

    att_kernel<<<FDIM / 256, 256, 0, stream>>>(emb_pad, W_att, b_att, xg);

    gat_kernel<<<1, NNODE, 0, stream>>>(xg, Wl0, bl0, Wr0, br0, a0, gb0, g0);
    gat_kernel<<<1, NNODE, 0, stream>>>(g0, Wl1, bl1, Wr1, br1, a1, gb1, g1);

    fuse_kernel<<<(NNODE * 32 + 255) / 256, 256, 0, stream>>>(xg, g0, g1, fusion);

    // priors GEMM: 1024 waves (4 waves/block)
    priors_kernel<<<256, 128, 0, stream>>>(W_caps, fusion, priors);

    routing_kernel<<<1, 256, 0, stream>>>(priors, W_fus, b_fus, logits, pbuf, outp);
}